// Shift_gcn_student_32186484916746
// MI455X (gfx1250) — compile-verified
//
#include <hip/hip_runtime.h>

// Shapes
#define N_  64
#define C_  64
#define T_  256
#define V_  25
#define D_  64
#define TV_ 6400       // T*V  (GEMM M per sample)
#define VD_ 1600       // V*D  (BN feature count)
#define NT_ 16384      // N*T  (BN sample count)
#define BN_EPS 1e-5f
#define SXPITCH 65     // padded LDS pitch for A tile (avoids stride-64 bank conflicts)

typedef __attribute__((ext_vector_type(2))) float v2f;
typedef __attribute__((ext_vector_type(8))) float v8f;
typedef __attribute__((ext_vector_type(4))) int   v4i;

// pointer-to-addrspace-qualified-v4i types for the async builtin
typedef __attribute__((address_space(1))) v4i* as1_v4i_p;   // global
typedef __attribute__((address_space(3))) v4i* as3_v4i_p;   // LDS

// ---- CDNA5 async global->LDS copy (ASYNCcnt path), with portable fallback ----
#if defined(__AMDGCN__) && __has_builtin(__builtin_amdgcn_global_load_async_to_lds_b128)
#define HAS_ASYNC_LDS 1
#else
#define HAS_ASYNC_LDS 0
#endif

__device__ __forceinline__ void async_copy_b128(const float* g, float* l) {
#if HAS_ASYNC_LDS
  __builtin_amdgcn_global_load_async_to_lds_b128(
      (as1_v4i_p)(void*)g, (as3_v4i_p)(void*)l, 0, 0);
#else
  float4 tmp = *(const float4*)g;
  *(float4*)l = tmp;
#endif
}

__device__ __forceinline__ void wait_async0() {
#if HAS_ASYNC_LDS
#if __has_builtin(__builtin_amdgcn_s_wait_asynccnt)
  __builtin_amdgcn_s_wait_asynccnt(0);
#else
  asm volatile("s_wait_asynccnt 0x0" ::: "memory");
#endif
#endif
}

// ---------------- workspace layout (in floats) ----------------
#define OFF_POOLED 0                         // 4096
#define OFF_MASK   (OFF_POOLED + 4096)       // 1600
#define OFF_WF     (OFF_MASK   + 1600)       // 64*4096 = 262144  (16B aligned)
#define OFF_BF     (OFF_WF     + 262144)     // 4096
#define OFF_SUM    (OFF_BF     + 4096)       // 1600
#define OFF_SUMSQ  (OFF_SUM    + 1600)       // 1600
#define OFF_MEAN   (OFF_SUMSQ  + 1600)       // 1600
#define OFF_RSTD   (OFF_MEAN   + 1600)       // 1600
#define OFF_Y      (OFF_RSTD   + 1600)       // 64*6400*64 = 26214400 (16B aligned)

// ---------------- K0: feature mask = tanh(FM)+1 ----------------
__global__ __launch_bounds__(256) void k_mask(const float* __restrict__ FM,
                                              float* __restrict__ maskt) {
  int i = blockIdx.x * 256 + threadIdx.x;
  if (i < VD_) maskt[i] = tanhf(FM[i]) + 1.0f;
}

// ---------------- K1: global average pool over (T,V) ----------------
__global__ __launch_bounds__(256) void k_pool(const float* __restrict__ x0,
                                              float* __restrict__ pooled) {
  __shared__ float red[256];
  const int b = blockIdx.x;                 // n*64 + c
  const int tid = threadIdx.x;
  const float* xb = x0 + (size_t)b * TV_;
  float s = 0.0f;
  for (int k = tid; k < TV_; k += 256) s += xb[k];
  red[tid] = s;
  __syncthreads();
  for (int off = 128; off > 0; off >>= 1) {
    if (tid < off) red[tid] += red[tid + off];
    __syncthreads();
  }
  if (tid == 0) pooled[b] = red[0] * (1.0f / (float)TV_);
}

// ---------------- K2: gate MLP + softmax + fused W/b per sample ----------------
__global__ __launch_bounds__(64) void k_gate(const float* __restrict__ pooled,
                                             const float* __restrict__ fc1w,
                                             const float* __restrict__ fc1b,
                                             const float* __restrict__ fc2w,
                                             const float* __restrict__ fc2b,
                                             const float* __restrict__ Lw,
                                             const float* __restrict__ Lb,
                                             const int*   __restrict__ epoch,
                                             float* __restrict__ Wf,
                                             float* __restrict__ bf) {
  __shared__ float p[64], hh[16], gt[4];
  const int n = blockIdx.x;
  const int tid = threadIdx.x;
  p[tid] = pooled[n * 64 + tid];
  __syncthreads();
  if (tid < 16) {
    float s = fc1b[tid];
    for (int c = 0; c < 64; ++c) s += p[c] * fc1w[tid * 64 + c];
    hh[tid] = fmaxf(s, 0.0f);
  }
  __syncthreads();
  if (tid == 0) {
    float z[4];
    for (int k = 0; k < 4; ++k) {
      float s = fc2b[k];
      for (int j = 0; j < 16; ++j) s += hh[j] * fc2w[k * 16 + j];
      z[k] = s;
    }
    int ep = epoch[0];
    float tao = (ep < 60) ? (-(29.0f / 60.0f) * (float)ep + 30.0f) : 1.0f;
    float inv = 1.0f / tao;
    float mx = z[0] * inv;
    for (int k = 0; k < 4; ++k) { z[k] *= inv; mx = fmaxf(mx, z[k]); }
    float sm = 0.0f;
    for (int k = 0; k < 4; ++k) { z[k] = expf(z[k] - mx); sm += z[k]; }
    float isum = 1.0f / sm;
    for (int k = 0; k < 4; ++k) gt[k] = z[k] * isum;
  }
  __syncthreads();
  const float g0 = gt[0], g1 = gt[1], g2 = gt[2], g3 = gt[3];
  float* Wn = Wf + (size_t)n * 4096;
  for (int c = 0; c < 64; ++c) {
    int o = c * 64 + tid;
    Wn[o] = g0 * Lw[o] + g1 * Lw[4096 + o] + g2 * Lw[8192 + o] + g3 * Lw[12288 + o];
  }
  bf[n * 64 + tid] = g0 * Lb[tid] + g1 * Lb[64 + tid] + g2 * Lb[128 + tid] + g3 * Lb[192 + tid];
}

// ---------------- K3: batched GEMM with shift-in gather, fp32 WMMA ----------------
// grid = (50, 64): blockIdx.x = 128-row M-tile, blockIdx.y = n. 256 threads = 8 waves.
// Each wave computes a 16(M) x 64(N) strip: 4 accumulators, 16 K-steps of K=4.
__global__ __launch_bounds__(256) void k_gemm(const float* __restrict__ x0,
                                              const float* __restrict__ Wf,
                                              const float* __restrict__ bf,
                                              const float* __restrict__ maskt,
                                              float* __restrict__ y) {
  __shared__ __align__(16) float sW[64 * 64];       // W_fuse[n]  (B matrix [c][d])
  __shared__ __align__(16) float sX[128 * SXPITCH]; // gathered+masked A tile [row][c], padded
  const int n  = blockIdx.y;
  const int m0 = blockIdx.x * 128;
  const int tid = threadIdx.x;

  // async DMA of W_fuse[n] into LDS (contiguous 16KB) — overlaps with the gather below
  const float* Wn = Wf + (size_t)n * 4096;
  for (int i = tid * 4; i < 4096; i += 1024) async_copy_b128(Wn + i, &sW[i]);

  // A[(t,v),c] = x0[n, c, t, (v+c)%25] * maskt[v*64+c]
  // c slow / row fast: adjacent lanes walk consecutive m -> contiguous 25-float
  // windows of x0[n,c,t,*] -> ~2 cache lines per 32-lane request instead of 32.
  const float* xn = x0 + (size_t)n * C_ * TV_;
  for (int e = tid; e < 128 * 64; e += 256) {
    int c   = e >> 7;        // 0..63
    int row = e & 127;       // 0..127
    int m = m0 + row;
    int t = m / 25;
    int v = m - t * 25;
    int vv = v + c; vv -= (vv / 25) * 25;
    sX[row * SXPITCH + c] = xn[(c * T_ + t) * V_ + vv] * maskt[v * 64 + c];
  }
  wait_async0();
  __syncthreads();

  const int lane = tid & 31;
  const int wv   = tid >> 5;
  const int l16  = lane & 15;
  const int half = lane >> 4;          // 0: K pair {0,1}, 1: K pair {2,3}
  const int arow = (wv << 4) + l16;    // A row for this lane

  v8f acc[4] = {};
  #pragma unroll
  for (int kk = 0; kk < 16; ++kk) {
    const int c0 = kk * 4 + half * 2;
    v2f a;
    a.x = sX[arow * SXPITCH + c0];
    a.y = sX[arow * SXPITCH + c0 + 1];
    #pragma unroll
    for (int ntile = 0; ntile < 4; ++ntile) {
      const int ncol = ntile * 16 + l16;
      v2f b;
      b.x = sW[c0 * 64 + ncol];
      b.y = sW[(c0 + 1) * 64 + ncol];
      acc[ntile] = __builtin_amdgcn_wmma_f32_16x16x4_f32(
          /*neg_a=*/false, a, /*neg_b=*/false, b,
          /*c_mod=*/(short)0, acc[ntile], /*reuse_a=*/false, /*reuse_b=*/false);
    }
  }

  // bias + store: C/D layout -> VGPR r: lanes0-15 M=r, lanes16-31 M=r+8; N=lane%16
  float* yn = y + ((size_t)n * TV_ + m0) * 64;
  const float* bfn = bf + n * 64;
  #pragma unroll
  for (int ntile = 0; ntile < 4; ++ntile) {
    const int ncol = ntile * 16 + l16;
    const float bias = bfn[ncol];
    #pragma unroll
    for (int r = 0; r < 8; ++r) {
      const int row = (wv << 4) + r + (half << 3);
      yn[(size_t)row * 64 + ncol] = acc[ntile][r] + bias;
    }
  }
}

// ---------------- K4a: zero BN accumulators ----------------
__global__ __launch_bounds__(256) void k_zero(float* __restrict__ p, int count) {
  int i = blockIdx.x * 256 + threadIdx.x;
  if (i < count) p[i] = 0.0f;
}

// ---------------- K4: per-(v,d) sum / sumsq of y over (n,t) ----------------
__global__ __launch_bounds__(256) void k_bnstats(const float* __restrict__ y,
                                                 float* __restrict__ sum,
                                                 float* __restrict__ sumsq) {
  __shared__ float rs[256], rq[256];
  const int v = blockIdx.x;    // 0..24
  const int n = blockIdx.y;    // 0..63
  const int tid = threadIdx.x;
  const int d = tid & 63;
  const int g = tid >> 6;
  const float* yb = y + (size_t)n * (TV_ * 64) + v * 64 + d;
  float s = 0.0f, q = 0.0f;
  for (int t = g; t < T_; t += 4) {
    float x = yb[(size_t)t * VD_];
    s += x; q += x * x;
  }
  rs[tid] = s; rq[tid] = q;
  __syncthreads();
  if (g == 0) {
    s = rs[d] + rs[64 + d] + rs[128 + d] + rs[192 + d];
    q = rq[d] + rq[64 + d] + rq[128 + d] + rq[192 + d];
    atomicAdd(&sum[v * 64 + d], s);
    atomicAdd(&sumsq[v * 64 + d], q);
  }
}

// ---------------- K5: mean / rstd ----------------
__global__ __launch_bounds__(256) void k_bnfinal(const float* __restrict__ sum,
                                                 const float* __restrict__ sumsq,
                                                 float* __restrict__ mean,
                                                 float* __restrict__ rstd) {
  int i = blockIdx.x * 256 + threadIdx.x;
  if (i < VD_) {
    float m = sum[i] * (1.0f / (float)NT_);
    float v = sumsq[i] * (1.0f / (float)NT_) - m * m;
    mean[i] = m;
    rstd[i] = rsqrtf(v + BN_EPS);
  }
}

// ---------------- K6: shift-out + BN + transpose + residual + ReLU ----------------
// grid = (64, 64): blockIdx.x = 4-t chunk, blockIdx.y = n.
// out[n, j, t, i] = relu( (y[n,t,(i-j)%25,j]-mean)*rstd*gamma[i*64+j]+beta[i*64+j] + x0[n,j,t,i] )
__global__ __launch_bounds__(256) void k_final(const float* __restrict__ y,
                                               const float* __restrict__ x0,
                                               const float* __restrict__ mean,
                                               const float* __restrict__ rstd,
                                               const float* __restrict__ gamma,
                                               const float* __restrict__ beta,
                                               float* __restrict__ out) {
  __shared__ __align__(16) float sy[4 * VD_];
  __shared__ float sm[VD_], sr[VD_], sg[VD_], sb[VD_];
  const int n  = blockIdx.y;
  const int t0 = blockIdx.x * 4;
  const int tid = threadIdx.x;
  const float* yb = y + (size_t)n * (TV_ * 64) + (size_t)t0 * VD_;
  // async DMA of the contiguous 25.6KB y tile into LDS
  for (int i = tid; i < 1600; i += 256) async_copy_b128(yb + i * 4, &sy[i * 4]);
  for (int i = tid; i < VD_; i += 256) {
    sm[i] = mean[i]; sr[i] = rstd[i]; sg[i] = gamma[i]; sb[i] = beta[i];
  }
  wait_async0();
  __syncthreads();
  for (int p = tid; p < 6400; p += 256) {
    int j  = p / 100;            // output channel d
    int q  = p - j * 100;
    int tl = q / 25;             // local t
    int i  = q - tl * 25;        // output vertex v
    int vp = i - j + 75; vp -= (vp / 25) * 25;   // (i-j) mod 25
    int fy = vp * 64 + j;        // y-side feature (stats index)
    int fg = i * 64 + j;         // gathered feature (gamma/beta index)
    float val = sy[tl * VD_ + fy];
    float o = (val - sm[fy]) * sr[fy] * sg[fg] + sb[fg];
    size_t oi = ((size_t)(n * 64 + j) * T_ + (t0 + tl)) * V_ + i;
    out[oi] = fmaxf(o + x0[oi], 0.0f);
  }
}

// ---------------- host launcher ----------------
extern "C" void kernel_launch(void* const* d_in, const int* in_sizes, int n_in,
                              void* d_out, int out_size, void* d_ws, size_t ws_size,
                              hipStream_t stream) {
  const float* x0    = (const float*)d_in[0];
  const float* fc1w  = (const float*)d_in[1];
  const float* fc1b  = (const float*)d_in[2];
  const float* fc2w  = (const float*)d_in[3];
  const float* fc2b  = (const float*)d_in[4];
  const float* Lw    = (const float*)d_in[5];
  const float* Lb    = (const float*)d_in[6];
  const float* FM    = (const float*)d_in[7];
  const float* gamma = (const float*)d_in[8];
  const float* beta  = (const float*)d_in[9];
  const int*   epoch = (const int*)d_in[12];
  float* out = (float*)d_out;
  float* ws  = (float*)d_ws;

  k_mask<<<(VD_ + 255) / 256, 256, 0, stream>>>(FM, ws + OFF_MASK);
  k_pool<<<N_ * C_, 256, 0, stream>>>(x0, ws + OFF_POOLED);
  k_gate<<<N_, 64, 0, stream>>>(ws + OFF_POOLED, fc1w, fc1b, fc2w, fc2b,
                                Lw, Lb, epoch, ws + OFF_WF, ws + OFF_BF);
  k_gemm<<<dim3(TV_ / 128, N_), 256, 0, stream>>>(x0, ws + OFF_WF, ws + OFF_BF,
                                                  ws + OFF_MASK, ws + OFF_Y);
  k_zero<<<(2 * VD_ + 255) / 256, 256, 0, stream>>>(ws + OFF_SUM, 2 * VD_);
  k_bnstats<<<dim3(V_, N_), 256, 0, stream>>>(ws + OFF_Y, ws + OFF_SUM, ws + OFF_SUMSQ);
  k_bnfinal<<<(VD_ + 255) / 256, 256, 0, stream>>>(ws + OFF_SUM, ws + OFF_SUMSQ,
                                                   ws + OFF_MEAN, ws + OFF_RSTD);
  k_final<<<dim3(T_ / 4, N_), 256, 0, stream>>>(ws + OFF_Y, x0, ws + OFF_MEAN,
                                                ws + OFF_RSTD, gamma, beta, out);
}